// SelfAttention_82489141887493
// MI455X (gfx1250) — compile-verified
//
#include <hip/hip_runtime.h>
#include <hip/hip_bf16.h>
#include <math.h>

// ---------------------------------------------------------------------------
// Causal MHA with QK-LayerNorm for MI455X (gfx1250), fp32 end-to-end using
// the native fp32 matrix op V_WMMA_F32_16X16X4_F32 (wave32, 16x16 C tile).
//
// B=2, T=3072, E=512, H=8, S=64.  Flash-attention style: scores never touch
// HBM; Q/K/V/ctx intermediates live in d_ws (4 * 6144*512 floats = 50.3 MB,
// fully L2-resident on the 192 MB L2).
//
// GEMM kernels are register-blocked 32x64 per wave: 2 A-frags x 4 B-frags ->
// 8 WMMAs per 6 b64 loads per k-step (0.75 loads/WMMA), ~2.7x the compute
// intensity of naive 16x16-per-wave tiling.
// ---------------------------------------------------------------------------

#define BATCH 2
#define SEQ   3072
#define EMB   512
#define HEADS 8
#define HDIM  64
#define BT    (BATCH * SEQ)       // 6144 rows

typedef __attribute__((ext_vector_type(2))) float v2f;
typedef __attribute__((ext_vector_type(8))) float v8f;

// wmma helper: D = A(16x4) * B(4x16) + C   (fp32, wave32)
__device__ __forceinline__ v8f wmma_f32(v2f a, v2f b, v8f c) {
  return __builtin_amdgcn_wmma_f32_16x16x4_f32(
      /*neg_a=*/false, a, /*neg_b=*/false, b,
      /*c_mod=*/(short)0, c, /*reuse_a=*/false, /*reuse_b=*/false);
}

// ---------------------------------------------------------------------------
// Kernel 1: q/k/v = x @ W^T.  One wave -> one 32x64 output tile.
//   grid.x = BT/32 = 192 row units, grid.y = tensor (q|k|v),
//   wave (0..7)   = 64-col unit within EMB=512.
// A-fragment: lane m = M row of x, VGPR pair = K {+0,+1}/{+2,+3} per
// lane-half -> contiguous float2 loads. B-fragment: lane n = row of W
// (B = W^T), same K split.
// ---------------------------------------------------------------------------
__global__ __launch_bounds__(256) void qkv_gemm_kernel(
    const float* __restrict__ x,  const float* __restrict__ Wq,
    const float* __restrict__ Wk, const float* __restrict__ Wv,
    float* __restrict__ qb, float* __restrict__ kb, float* __restrict__ vb) {
  const int lane  = threadIdx.x & 31;
  const int wave  = threadIdx.x >> 5;     // 64-col unit
  const int ru    = blockIdx.x;           // 32-row unit
  const int which = blockIdx.y;           // 0:q 1:k 2:v

  const float* W   = (which == 0) ? Wq : (which == 1) ? Wk : Wv;
  float*       out = (which == 0) ? qb : (which == 1) ? kb : vb;

  const int m    = lane & 15;
  const int koff = (lane >> 4) << 1;      // 0 or 2
  const float* a0 = x + (size_t)(ru * 32 + m) * EMB + koff;
  const float* a1 = a0 + (size_t)16 * EMB;
  const float* b0 = W + (size_t)(wave * 64 + m) * EMB + koff;

  v8f acc[8] = {};                        // [nt] rows 0-15, [4+nt] rows 16-31
#pragma unroll 2
  for (int kk = 0; kk < EMB / 4; ++kk) {
    v2f af0 = *(const v2f*)(a0 + kk * 4);
    v2f af1 = *(const v2f*)(a1 + kk * 4);
#pragma unroll
    for (int nt = 0; nt < 4; ++nt) {
      v2f bf = *(const v2f*)(b0 + (size_t)nt * 16 * EMB + kk * 4);
      acc[nt]     = wmma_f32(af0, bf, acc[nt]);
      acc[4 + nt] = wmma_f32(af1, bf, acc[4 + nt]);
    }
  }
  // C layout: vgpr j -> row (j + 8*laneHalf), lane&15 -> col
  const int rbase = ru * 32 + (lane >> 4) * 8;
  const int colb  = wave * 64 + m;
#pragma unroll
  for (int nt = 0; nt < 4; ++nt)
#pragma unroll
    for (int j = 0; j < 8; ++j) {
      out[(size_t)(rbase + j) * EMB + colb + nt * 16]      = acc[nt][j];
      out[(size_t)(rbase + 16 + j) * EMB + colb + nt * 16] = acc[4 + nt][j];
    }
}

// ---------------------------------------------------------------------------
// Kernel 2: in-place LayerNorm over head dim (64) for q and k.
// One wave per (tensor, row, head): 2 elems/lane, wave32 shfl_xor reductions.
// ---------------------------------------------------------------------------
__global__ __launch_bounds__(256) void ln_qk_kernel(
    float* __restrict__ qb, float* __restrict__ kb,
    const float* __restrict__ qg, const float* __restrict__ qbeta,
    const float* __restrict__ kg, const float* __restrict__ kbeta) {
  const int gw   = (blockIdx.x * 256 + threadIdx.x) >> 5;  // 0 .. 2*BT*HEADS-1
  const int lane = threadIdx.x & 31;
  const int tens = (gw >= BT * HEADS) ? 1 : 0;
  const int r    = tens ? (gw - BT * HEADS) : gw;
  float* base = (tens ? kb : qb) + (size_t)(r >> 3) * EMB + (r & 7) * HDIM;
  const float* g = tens ? kg : qg;
  const float* b = tens ? kbeta : qbeta;

  v2f xv = *(const v2f*)(base + lane * 2);
  float s = xv.x + xv.y;
#pragma unroll
  for (int d = 1; d < 32; d <<= 1) s += __shfl_xor(s, d, 32);
  const float mu = s * (1.0f / 64.0f);
  const float d0 = xv.x - mu, d1 = xv.y - mu;
  float vs = d0 * d0 + d1 * d1;
#pragma unroll
  for (int d = 1; d < 32; d <<= 1) vs += __shfl_xor(vs, d, 32);
  const float rs = rsqrtf(vs * (1.0f / 64.0f) + 1e-5f);
  v2f y;
  y.x = d0 * rs * g[lane * 2]     + b[lane * 2];
  y.y = d1 * rs * g[lane * 2 + 1] + b[lane * 2 + 1];
  *(v2f*)(base + lane * 2) = y;
}

// ---------------------------------------------------------------------------
// Kernel 3: flash attention. One wave per (batch, head, 16-row query tile).
// S(16x16) = Q(16x64) Kt(64x16) via 16 WMMAs; online softmax with 16-lane
// (half-wave) row reductions; P re-laid C->A through a 1 KB per-wave LDS
// patch; O(16x64) += P V via 16 WMMAs per key tile.
// ---------------------------------------------------------------------------
__global__ __launch_bounds__(256) void attn_kernel(
    const float* __restrict__ qb, const float* __restrict__ kb,
    const float* __restrict__ vb, float* __restrict__ ctx) {
  __shared__ float pbuf[8][16 * 16];
  const int lane = threadIdx.x & 31;
  const int wave = threadIdx.x >> 5;
  const int wid  = blockIdx.x * 8 + wave;      // 0 .. 3071 (exact)
  const int qt    = wid % (SEQ / 16);          // query tile
  const int bh    = wid / (SEQ / 16);
  const int batch = bh >> 3;
  const int head  = bh & 7;
  const size_t rowbase = (size_t)batch * SEQ;

  const int m    = lane & 15;
  const int half = lane >> 4;
  const int koff = half * 2;

  // Cache Q fragments for the whole tile: 16 K-steps over head dim 64.
  const float* qrow = qb + (rowbase + qt * 16 + m) * EMB + head * HDIM + koff;
  v2f qa[16];
#pragma unroll
  for (int kk = 0; kk < 16; ++kk) qa[kk] = *(const v2f*)(qrow + kk * 4);

  v8f o0 = {}, o1 = {}, o2 = {}, o3 = {};
  float rowm[8], rowl[8];
#pragma unroll
  for (int j = 0; j < 8; ++j) { rowm[j] = -__builtin_inff(); rowl[j] = 0.f; }

  float* pl = pbuf[wave];

  for (int kt = 0; kt <= qt; ++kt) {
    // ---- S = Q @ K^T : B-fragment lane n = key row of K, contiguous d.
    const float* krow = kb + (rowbase + kt * 16 + m) * EMB + head * HDIM + koff;
    v8f s = {};
#pragma unroll
    for (int kk = 0; kk < 16; ++kk) {
      v2f kf = *(const v2f*)(krow + kk * 4);
      s = wmma_f32(qa[kk], kf, s);
    }
    // ---- online softmax (scale 1/sqrt(64) = 0.125)
    const bool diag = (kt == qt);
    const int  kcol = kt * 16 + m;
#pragma unroll
    for (int j = 0; j < 8; ++j) {
      const int qrowi = qt * 16 + j + half * 8;
      float sv = s[j] * 0.125f;
      if (diag && kcol > qrowi) sv = -__builtin_inff();
      float mx = sv;
#pragma unroll
      for (int d = 1; d < 16; d <<= 1) mx = fmaxf(mx, __shfl_xor(mx, d, 32));
      const float mnew = fmaxf(rowm[j], mx);
      const float p    = __expf(sv - mnew);
      float ps = p;
#pragma unroll
      for (int d = 1; d < 16; d <<= 1) ps += __shfl_xor(ps, d, 32);
      const float corr = __expf(rowm[j] - mnew);
      rowl[j] = rowl[j] * corr + ps;
      rowm[j] = mnew;
      o0[j] *= corr; o1[j] *= corr; o2[j] *= corr; o3[j] *= corr;
      pl[(j + half * 8) * 16 + m] = p;     // C-layout -> LDS (row-major 16x16)
    }
    __asm__ volatile("s_wait_dscnt 0" ::: "memory");  // wave-local LDS visibility
    // ---- O += P @ V : A-fragment of P from LDS, B-fragment of V strided.
    const float* vbase = vb + (rowbase + kt * 16) * EMB + head * HDIM;
#pragma unroll
    for (int kk = 0; kk < 4; ++kk) {
      v2f pa;
      pa.x = pl[m * 16 + kk * 4 + koff];
      pa.y = pl[m * 16 + kk * 4 + koff + 1];
      const float* v0 = vbase + (size_t)(kk * 4 + koff) * EMB;
      v2f bv;
      bv.x = v0[m];            bv.y = v0[EMB + m];
      o0 = wmma_f32(pa, bv, o0);
      bv.x = v0[16 + m];       bv.y = v0[EMB + 16 + m];
      o1 = wmma_f32(pa, bv, o1);
      bv.x = v0[32 + m];       bv.y = v0[EMB + 32 + m];
      o2 = wmma_f32(pa, bv, o2);
      bv.x = v0[48 + m];       bv.y = v0[EMB + 48 + m];
      o3 = wmma_f32(pa, bv, o3);
    }
  }
  // ---- normalize and store context (b,t,h,s) == row-major (BT, EMB)
  float* obase = ctx + (rowbase + qt * 16) * EMB + head * HDIM;
#pragma unroll
  for (int j = 0; j < 8; ++j) {
    const float inv = 1.0f / rowl[j];
    const int rr = j + half * 8;
    obase[(size_t)rr * EMB + m]      = o0[j] * inv;
    obase[(size_t)rr * EMB + 16 + m] = o1[j] * inv;
    obase[(size_t)rr * EMB + 32 + m] = o2[j] * inv;
    obase[(size_t)rr * EMB + 48 + m] = o3[j] * inv;
  }
}

// ---------------------------------------------------------------------------
// Kernel 4: out = ctx @ Wo^T + bo.  32x64 per wave, same blocking as kernel 1.
//   grid.x = BT/32 = 192 row units, wave = 64-col unit.
// ---------------------------------------------------------------------------
__global__ __launch_bounds__(256) void proj_kernel(
    const float* __restrict__ ctx, const float* __restrict__ Wo,
    const float* __restrict__ bo, float* __restrict__ out) {
  const int lane = threadIdx.x & 31;
  const int wave = threadIdx.x >> 5;
  const int ru   = blockIdx.x;
  const int m    = lane & 15;
  const int koff = (lane >> 4) << 1;
  const float* a0 = ctx + (size_t)(ru * 32 + m) * EMB + koff;
  const float* a1 = a0 + (size_t)16 * EMB;
  const float* b0 = Wo + (size_t)(wave * 64 + m) * EMB + koff;

  v8f acc[8] = {};
#pragma unroll 2
  for (int kk = 0; kk < EMB / 4; ++kk) {
    v2f af0 = *(const v2f*)(a0 + kk * 4);
    v2f af1 = *(const v2f*)(a1 + kk * 4);
#pragma unroll
    for (int nt = 0; nt < 4; ++nt) {
      v2f bf = *(const v2f*)(b0 + (size_t)nt * 16 * EMB + kk * 4);
      acc[nt]     = wmma_f32(af0, bf, acc[nt]);
      acc[4 + nt] = wmma_f32(af1, bf, acc[4 + nt]);
    }
  }
  const int rbase = ru * 32 + (lane >> 4) * 8;
  const int colb  = wave * 64 + m;
#pragma unroll
  for (int nt = 0; nt < 4; ++nt) {
    const float bias = bo[colb + nt * 16];
#pragma unroll
    for (int j = 0; j < 8; ++j) {
      out[(size_t)(rbase + j) * EMB + colb + nt * 16]      = acc[nt][j] + bias;
      out[(size_t)(rbase + 16 + j) * EMB + colb + nt * 16] = acc[4 + nt][j] + bias;
    }
  }
}

// ---------------------------------------------------------------------------
extern "C" void kernel_launch(void* const* d_in, const int* in_sizes, int n_in,
                              void* d_out, int out_size, void* d_ws, size_t ws_size,
                              hipStream_t stream) {
  (void)in_sizes; (void)n_in; (void)out_size; (void)ws_size;
  const float* x     = (const float*)d_in[0];
  const float* Wq    = (const float*)d_in[1];
  const float* Wk    = (const float*)d_in[2];
  const float* Wv    = (const float*)d_in[3];
  const float* Wo    = (const float*)d_in[4];
  const float* bo    = (const float*)d_in[5];
  const float* qg    = (const float*)d_in[6];
  const float* qbeta = (const float*)d_in[7];
  const float* kg    = (const float*)d_in[8];
  const float* kbeta = (const float*)d_in[9];

  float* ws = (float*)d_ws;
  const size_t per = (size_t)BT * EMB;   // 3,145,728 floats per tensor
  float* qbuf = ws;
  float* kbuf = ws + per;
  float* vbuf = ws + 2 * per;
  float* cbuf = ws + 3 * per;

  // 1) q/k/v projections: 192 row units x 8 col units x 3 tensors
  qkv_gemm_kernel<<<dim3(BT / 32, 3), 256, 0, stream>>>(x, Wq, Wk, Wv,
                                                        qbuf, kbuf, vbuf);
  // 2) QK-LayerNorm: 2 * 6144 * 8 waves, 8 waves/block
  ln_qk_kernel<<<(2 * BT * HEADS) / 8, 256, 0, stream>>>(qbuf, kbuf,
                                                         qg, qbeta, kg, kbeta);
  // 3) flash attention: 2*8*192 = 3072 waves, 8 waves/block
  attn_kernel<<<(BATCH * HEADS * (SEQ / 16)) / 8, 256, 0, stream>>>(
      qbuf, kbuf, vbuf, cbuf);
  // 4) output projection (+bias): 192 row units x 8 col units
  proj_kernel<<<BT / 32, 256, 0, stream>>>(cbuf, Wo, bo, (float*)d_out);
}